// WaveContrastiveLoss_72284299592263
// MI455X (gfx1250) — compile-verified
//
#include <hip/hip_runtime.h>
#include <math.h>

typedef __attribute__((ext_vector_type(2))) float v2f;
typedef __attribute__((ext_vector_type(8))) float v8f;

#define NWAVES    32      // waves per set (pos or neg)
#define NB        64      // batches
#define NT        8192    // samples per wave
#define TAU_F     0.08f
#define CHUNK     128     // t-samples staged per chunk
#define LDS_STR   132     // 128 + 4 pad floats: conflict-free rows, 16B-aligned (528B)
#define KSPLIT    4       // K-dimension split for occupancy
#define NCH       ((NT / CHUNK) / KSPLIT)   // 16 chunks per workgroup

// ---------------------------------------------------------------------------
// CDNA5 async global->LDS copy (ASYNCcnt-tracked, no VGPR round trip)
// ---------------------------------------------------------------------------
__device__ __forceinline__ void async_ld_b128(unsigned lds_byte_addr, const float* gaddr) {
  asm volatile("global_load_async_to_lds_b128 %0, %1, off"
               :: "v"(lds_byte_addr), "v"(gaddr) : "memory");
}

// ---------------------------------------------------------------------------
// Kernel 1: per (b, kpart) partial 64x64 Gram of [pos;neg] rows via f32 WMMA.
// Double-buffered LDS fed by async loads; s1 partials recomputed from LDS.
// Gp layout: [KSPLIT][NB][64][64] ; S1p layout: [KSPLIT][NB][64]
// ---------------------------------------------------------------------------
__global__ __launch_bounds__(512)
void WaveContrastive_gram_kernel(const float* __restrict__ pos,
                                 const float* __restrict__ neg,
                                 float* __restrict__ Gp,
                                 float* __restrict__ S1p) {
  __shared__ float As[2 * 64 * LDS_STR];

  const int part = blockIdx.x / NB;     // 0..KSPLIT-1
  const int b    = blockIdx.x % NB;     // 0..63
  const int wave = threadIdx.x >> 5;    // 0..15
  const int lane = threadIdx.x & 31;    // 0..31
  const int tbase = part * NCH * CHUNK;

  // Staging: wave w stages rows {w, w+16, w+32, w+48}; lane = float4 slot.
  const float* base[4];
#pragma unroll
  for (int i = 0; i < 4; ++i) {
    const int r = wave + 16 * i;
    const float* p = (r < NWAVES) ? pos : neg;
    base[i] = p + (size_t)(r & (NWAVES - 1)) * ((size_t)NB * NT)
                + (size_t)b * NT + tbase + lane * 4;
  }
  const unsigned As0 = (unsigned)(uintptr_t)As;   // low 32 bits = LDS byte offset

  auto issue = [&](int ck) {
    const int buf = ck & 1;
    const int t0  = ck * CHUNK;
#pragma unroll
    for (int i = 0; i < 4; ++i) {
      const int r = wave + 16 * i;
      const unsigned la = As0 + (unsigned)((((buf * 64 + r) * LDS_STR) + lane * 4) * 4);
      async_ld_b128(la, base[i] + t0);
    }
  };

  // WMMA tile assignment: wave w -> 16x16 tile (I,J) of the 64x64 Gram.
  const int I = wave >> 2;
  const int J = wave & 3;
  const int halfsel = lane >> 4;        // 0: K=0,1   1: K=2,3 (ISA 16x4 f32 layout)
  const int l15     = lane & 15;
  const int arow = I * 16 + l15;
  const int brow = J * 16 + l15;

  // s1 assignment: thread sums row (tid>>3), cols seg*16..seg*16+15
  const int srow = threadIdx.x >> 3;
  const int seg  = threadIdx.x & 7;

  v8f c = {0.f, 0.f, 0.f, 0.f, 0.f, 0.f, 0.f, 0.f};
  float ssum = 0.f;

  issue(0);
  issue(1);

#pragma unroll 1
  for (int ck = 0; ck < NCH; ++ck) {
    // Wait for this chunk's own 4 async ops (in-order); next chunk may still fly.
    if (ck < NCH - 1) asm volatile("s_wait_asynccnt 0x4" ::: "memory");
    else              asm volatile("s_wait_asynccnt 0x0" ::: "memory");
    __syncthreads();                       // all waves' chunk-ck data visible

    const float* Ab = &As[(ck & 1) * 64 * LDS_STR];

#pragma unroll
    for (int kk = 0; kk < CHUNK / 4; ++kk) {
      const int col = kk * 4 + halfsel * 2;
      v2f a  = *(const v2f*)&Ab[arow * LDS_STR + col];
      v2f bb = *(const v2f*)&Ab[brow * LDS_STR + col];
      c = __builtin_amdgcn_wmma_f32_16x16x4_f32(false, a, false, bb,
                                                (short)0, c, false, false);
    }

    // s1 partial from the staged chunk (zero extra HBM traffic)
#pragma unroll
    for (int j = 0; j < 4; ++j) {
      const float4 v = *(const float4*)&Ab[srow * LDS_STR + seg * 16 + j * 4];
      ssum += v.x + v.y + v.z + v.w;
    }

    __syncthreads();                       // everyone done reading buf (ck&1)
    if (ck + 2 < NCH) issue(ck + 2);       // refill the buffer just freed
  }

  // Reduce s1 across the 8 threads sharing a row (fixed order -> deterministic)
  ssum += __shfl_xor(ssum, 1, 32);
  ssum += __shfl_xor(ssum, 2, 32);
  ssum += __shfl_xor(ssum, 4, 32);
  if (seg == 0) S1p[((size_t)part * NB + b) * 64 + srow] = ssum;

  // C/D layout: VGPR r, lanes 0-15: M=r, N=lane; lanes 16-31: M=r+8, N=lane-16
  const int mBase = I * 16 + halfsel * 8;
  const int nCol  = J * 16 + l15;
#pragma unroll
  for (int r = 0; r < 8; ++r) {
    Gp[(((size_t)part * NB + b) * 64 + (mBase + r)) * 64 + nCol] = c[r];
  }
}

// ---------------------------------------------------------------------------
// Kernel 2: per (n,m) pair -> sum partials -> mean_b(1 - pearson) -> exp(d/tau)
// ---------------------------------------------------------------------------
__global__ void WaveContrastive_pair_kernel(const float* __restrict__ Gp,
                                            const float* __restrict__ S1p,
                                            float* __restrict__ expPos,
                                            float* __restrict__ expNeg) {
  const int gid = blockIdx.x * blockDim.x + threadIdx.x;  // 0..4095
  const int n = gid >> 6;
  const int m = gid & 63;

  float pv = 0.f, nv = 0.f;
  const bool isPos = (n < NWAVES) && (m < NWAVES) && (n < m);   // upper tri of PxP
  const bool isNeg = (n < NWAVES) && (m >= NWAVES);             // PxQ
  if (isPos || isNeg) {
    const float T = (float)NT;
    float dsum = 0.f;
    for (int b = 0; b < NB; ++b) {
      float g = 0.f, snn = 0.f, smm = 0.f, s1n = 0.f, s1m = 0.f;
#pragma unroll
      for (int p = 0; p < KSPLIT; ++p) {
        const size_t gb = ((size_t)p * NB + b) * 64;
        g   += Gp[(gb + n) * 64 + m];
        snn += Gp[(gb + n) * 64 + n];
        smm += Gp[(gb + m) * 64 + m];
        s1n += S1p[gb + n];
        s1m += S1p[gb + m];
      }
      const float num = T * g - s1n * s1m;
      const float den = sqrtf((T * snn - s1n * s1n) * (T * smm - s1m * s1m));
      dsum += 1.f - num / den;
    }
    const float val = expf((dsum * (1.f / (float)NB)) / TAU_F);
    if (isPos) pv = val; else nv = val;
  }
  expPos[gid] = pv;
  expNeg[gid] = nv;
}

// ---------------------------------------------------------------------------
// Kernel 3: deterministic fixed-order reduction + final scalar
// ---------------------------------------------------------------------------
__global__ void WaveContrastive_finalize_kernel(const float* __restrict__ expPos,
                                                const float* __restrict__ expNeg,
                                                float* __restrict__ out) {
  __shared__ float sp[256];
  __shared__ float sn[256];
  const int tid = threadIdx.x;
  float p = 0.f, q = 0.f;
  for (int i = tid; i < 4096; i += 256) { p += expPos[i]; q += expNeg[i]; }
  sp[tid] = p; sn[tid] = q;
  __syncthreads();
  for (int s = 128; s > 0; s >>= 1) {
    if (tid < s) { sp[tid] += sp[tid + s]; sn[tid] += sn[tid + s]; }
    __syncthreads();
  }
  if (tid == 0) out[0] = log10f(sp[0] / sn[0] + 1.f);
}

// ---------------------------------------------------------------------------
extern "C" void kernel_launch(void* const* d_in, const int* in_sizes, int n_in,
                              void* d_out, int out_size, void* d_ws, size_t ws_size,
                              hipStream_t stream) {
  (void)in_sizes; (void)n_in; (void)out_size; (void)ws_size;
  const float* pos = (const float*)d_in[0];
  const float* neg = (const float*)d_in[1];

  float* Gp  = (float*)d_ws;                      // KSPLIT*64*64*64 f32 = 4 MB
  float* S1p = Gp + (size_t)KSPLIT * NB * 64 * 64; // KSPLIT*64*64 f32
  float* eP  = S1p + (size_t)KSPLIT * NB * 64;     // 4096 f32
  float* eN  = eP + 4096;                          // 4096 f32

  WaveContrastive_gram_kernel<<<NB * KSPLIT, 512, 0, stream>>>(pos, neg, Gp, S1p);
  WaveContrastive_pair_kernel<<<16, 256, 0, stream>>>(Gp, S1p, eP, eN);
  WaveContrastive_finalize_kernel<<<1, 256, 0, stream>>>(eP, eN, (float*)d_out);
}